// TestLoss_88699664597867
// MI455X (gfx1250) — compile-verified
//
#include <hip/hip_runtime.h>
#include <hip/hip_bf16.h>

typedef float v2f __attribute__((ext_vector_type(2)));
typedef float v8f __attribute__((ext_vector_type(8)));

#define B_   4
#define N_   8192
#define M_   8192

// ---------------------------------------------------------------------------
// Prep: for each 3D point build
//   A-form  (p0, p1, p2, |p|^2)   -- used when this set is the "query" rows
//   B-form  (-2p0, -2p1, -2p2, 1) -- used when this set is the "target" cols
//   nrm     |p|^2                 -- per-column constant added in epilogue
// ---------------------------------------------------------------------------
__global__ void chamfer_prep(const float* __restrict__ pts, int count,
                             float* __restrict__ Aout,
                             float* __restrict__ Bout,
                             float* __restrict__ nrm) {
  int i = blockIdx.x * blockDim.x + threadIdx.x;
  if (i >= count) return;
  float x = pts[3 * i + 0];
  float y = pts[3 * i + 1];
  float z = pts[3 * i + 2];
  float n2 = x * x + y * y + z * z;
  Aout[4 * i + 0] = x;      Aout[4 * i + 1] = y;
  Aout[4 * i + 2] = z;      Aout[4 * i + 3] = n2;
  Bout[4 * i + 0] = -2.0f * x;  Bout[4 * i + 1] = -2.0f * y;
  Bout[4 * i + 2] = -2.0f * z;  Bout[4 * i + 3] = 1.0f;
  nrm[i] = n2;
}

// ---------------------------------------------------------------------------
// For every query point p (rows), min over all target points q (cols) of
// |p-q|^2, via V_WMMA_F32_16X16X4_F32 tiles.
// Each wave (32 lanes) owns 16 rows; 8 waves/block -> 128 rows per block.
// grid = (NP/128, B). Fully deterministic (no atomics).
// ---------------------------------------------------------------------------
__global__ void __launch_bounds__(256)
chamfer_nn_min(const float* __restrict__ PA,   // [B*NP*4] query A-form
               const float* __restrict__ QB,   // [B*NQ*4] target B-form
               const float* __restrict__ Q2,   // [B*NQ]   target |q|^2
               float* __restrict__ mins,       // [B*NP]   row mins out
               int NP, int NQ) {
  const int b    = blockIdx.y;
  const int lane = threadIdx.x & 31;
  const int wave = threadIdx.x >> 5;
  const int rowBase = blockIdx.x * 128 + wave * 16;

  const float* pa = PA + (size_t)b * NP * 4;
  const float* qb = QB + (size_t)b * NQ * 4;
  const float* q2 = Q2 + (size_t)b * NQ;

  // ISA 7.12.2, 32-bit 16x4 A-matrix: lane%16 = M (row), lanes 0-15 hold
  // K={0,1} in {v0,v1}, lanes 16-31 hold K={2,3}.  B (4x16) mirrors this
  // with lane%16 = N (col).  Both are one aligned float2 load per lane.
  const int rl = lane & 15;           // row (A) / col (B) within tile
  const int kb = (lane >> 4) * 2;     // K base for this lane half

  v2f afrag = *(const v2f*)(pa + (size_t)(rowBase + rl) * 4 + kb);

  float runmin[8];
#pragma unroll
  for (int r = 0; r < 8; ++r) runmin[r] = 3.0e38f;

#pragma unroll 2
  for (int q0 = 0; q0 < NQ; q0 += 16) {
    v2f bfrag = *(const v2f*)(qb + (size_t)(q0 + rl) * 4 + kb);
    float qn  = q2[q0 + rl];          // per-column |q|^2 (column == lane)

    v8f c = {};
    // (neg_a, A, neg_b, B, c_mod, C, reuse_a, reuse_b)
    c = __builtin_amdgcn_wmma_f32_16x16x4_f32(
        false, afrag, false, bfrag, (short)0, c, false, false);

    // c[r] = |p|^2 - 2 p.q  for row (r + 8*(lane/16)), col (lane%16)
#pragma unroll
    for (int r = 0; r < 8; ++r) {
      float d = c[r] + qn;
      runmin[r] = fminf(runmin[r], d);
    }
  }

  // Min across the 16 columns (lanes) within each lane-half, then store the
  // two rows this VGPR slot covers (M=r in lanes 0-15, M=r+8 in lanes 16-31).
#pragma unroll
  for (int r = 0; r < 8; ++r) {
    float v = runmin[r];
    v = fminf(v, __shfl_xor(v, 1, 32));
    v = fminf(v, __shfl_xor(v, 2, 32));
    v = fminf(v, __shfl_xor(v, 4, 32));
    v = fminf(v, __shfl_xor(v, 8, 32));
    if (rl == 0) {
      int row = rowBase + r + (lane >> 4) * 8;
      mins[(size_t)b * NP + row] = v;
    }
  }
}

// ---------------------------------------------------------------------------
// Deterministic single-block sum of all 2*B*N row-mins, scaled by 1/(B*N).
// ---------------------------------------------------------------------------
__global__ void chamfer_reduce(const float* __restrict__ mins, int count,
                               float scale, float* __restrict__ out) {
  __shared__ float sdata[256];
  float s = 0.0f;
  for (int i = threadIdx.x; i < count; i += 256) s += mins[i];
  sdata[threadIdx.x] = s;
  __syncthreads();
  for (int off = 128; off > 0; off >>= 1) {
    if ((int)threadIdx.x < off) sdata[threadIdx.x] += sdata[threadIdx.x + off];
    __syncthreads();
  }
  if (threadIdx.x == 0) out[0] = sdata[0] * scale;
}

extern "C" void kernel_launch(void* const* d_in, const int* in_sizes, int n_in,
                              void* d_out, int out_size, void* d_ws, size_t ws_size,
                              hipStream_t stream) {
  const float* res = (const float*)d_in[0];  // [B, N, 3]
  const float* gt  = (const float*)d_in[1];  // [B, M, 3]

  const int BN = B_ * N_;
  const int BM = B_ * M_;

  float* ws   = (float*)d_ws;
  float* resA = ws;                 // BN*4
  float* resB = resA + (size_t)BN * 4;
  float* res2 = resB + (size_t)BN * 4;  // BN
  float* gtA  = res2 + BN;          // BM*4
  float* gtB  = gtA + (size_t)BM * 4;
  float* gt2  = gtB + (size_t)BM * 4;   // BM
  float* mins = gt2 + BM;           // BN + BM

  chamfer_prep<<<(BN + 255) / 256, 256, 0, stream>>>(res, BN, resA, resB, res2);
  chamfer_prep<<<(BM + 255) / 256, 256, 0, stream>>>(gt,  BM, gtA,  gtB,  gt2);

  dim3 grid_x(N_ / 128, B_);
  // direction 1: for each res point, nearest gt
  chamfer_nn_min<<<grid_x, 256, 0, stream>>>(resA, gtB, gt2, mins, N_, M_);
  dim3 grid_y(M_ / 128, B_);
  // direction 2: for each gt point, nearest res
  chamfer_nn_min<<<grid_y, 256, 0, stream>>>(gtA, resB, res2, mins + BN, M_, N_);

  // mean_b(cham_x + cham_y) = (sum of all mins) / (B*N)   (N == M)
  chamfer_reduce<<<1, 256, 0, stream>>>(mins, BN + BM, 1.0f / (float)BN,
                                        (float*)d_out);
}